// CA_90958817395357
// MI455X (gfx1250) — compile-verified
//
#include <hip/hip_runtime.h>
#include <math.h>

// CA step: depthwise 5x5 (D4-symmetrized, zero-mean) + leaky -> 8x8 GEMM +
// leaky -> 8x8 GEMM + residual + tanh.
//  - pointwise GEMMs on v_wmma_f32_16x16x4_f32 (fp32, two ops cover K=8)
//  - psi tile staged with global_load_async_to_lds_b128 (ASYNCcnt path)

#define HIDDEN   8
#define RESOL    64
#define TILE_R   8
#define LROWS    (TILE_R + 4)            // 12 rows incl. 2-row halo each side
#define LCOLS    72                      // 2 pad | 2 halo | 64 interior | 2 halo | 2 pad (16B-aligned interior)
#define COL0     4                       // lds col of global col 0
#define PSI_PLANE (LROWS * LCOLS)        // 864 floats per channel plane
#define Z1_PLANE  (TILE_R * RESOL + 16)  // 528: pad so B-frag gathers avoid bank conflicts

typedef float v2f __attribute__((ext_vector_type(2)));
typedef float v8f __attribute__((ext_vector_type(8)));

__device__ __forceinline__ float leaky(float x) { return x > 0.0f ? x : 0.01f * x; }

__global__ __launch_bounds__(256)
void ca_step_kernel(const float* __restrict__ psi,
                    const float* __restrict__ filter1,
                    const float* __restrict__ bias1,
                    const float* __restrict__ w2,
                    const float* __restrict__ b2,
                    const float* __restrict__ w3,
                    const float* __restrict__ b3,
                    float* __restrict__ out)
{
    __shared__ float s_psi[HIDDEN * PSI_PLANE];  // 27648 B
    __shared__ float s_z1 [HIDDEN * Z1_PLANE];   // 16896 B
    __shared__ float s_w  [HIDDEN * 25];         //   800 B

    const int tid  = threadIdx.x;
    const int lane = tid & 31;
    const int wave = tid >> 5;
    const int b    = blockIdx.y;
    const int r0   = blockIdx.x * TILE_R;

    // ---- stage psi tile: async DMA rows into LDS, zero the halo ----
    // 96 (channel, lds-row) rows; 16-thread unit copies one 64-float row
    // (16 lanes x b128) straight into LDS, tracked on ASYNCcnt.
    {
        const int unit = tid >> 4;     // 0..15
        const int sub  = tid & 15;     // 0..15
        #pragma unroll
        for (int q = 0; q < 6; ++q) {
            const int row_id = unit * 6 + q;           // 0..95
            const int c  = row_id / LROWS;
            const int rr = row_id % LROWS;
            const int gr = r0 + rr - 2;
            const int rbase = c * PSI_PLANE + rr * LCOLS;
            if (gr >= 0 && gr < RESOL) {
                // side halo (global cols -2,-1,64,65) is always out of bounds -> 0
                if (sub < 2) {
                    const int hb = rbase + (sub ? (COL0 + RESOL) : (COL0 - 2));
                    s_psi[hb] = 0.0f; s_psi[hb + 1] = 0.0f;
                }
                const uint32_t ldsoff =
                    (uint32_t)(uintptr_t)(&s_psi[rbase + COL0 + sub * 4]);   // 16B aligned
                const uint64_t gaddr =
                    (uint64_t)(uintptr_t)(psi + (((size_t)(b * HIDDEN + c) * RESOL + gr) * RESOL
                                                 + sub * 4));
                asm volatile("global_load_async_to_lds_b128 %0, %1, off"
                             :: "v"(ldsoff), "v"(gaddr) : "memory");
            } else {
                // whole row out of bounds (image top/bottom) -> zero it
                for (int j = sub; j < LCOLS; j += 16) s_psi[rbase + j] = 0.0f;
            }
        }
    }

    // ---- D4-symmetrized, zero-mean 5x5 kernels (mean(z) == mean(x)) ----
    if (tid < HIDDEN * 25) {
        const int c = tid / 25, k = tid % 25;
        const int i = k / 5,   j = k % 5;
        const float* f = filter1 + c * 25;
        float z = 0.125f * (f[i*5+j]     + f[(4-i)*5+j]     + f[i*5+(4-j)]     + f[(4-i)*5+(4-j)]
                          + f[j*5+i]     + f[j*5+(4-i)]     + f[(4-j)*5+i]     + f[(4-j)*5+(4-i)]);
        float s = 0.0f;
        #pragma unroll
        for (int t = 0; t < 25; ++t) s += f[t];
        s_w[tid] = z - s * (1.0f / 25.0f);
    }

    // ---- WMMA A-fragments (w2/w3), 16x4 f32: rows 8..15 zeroed ----
    const int arow = lane & 15;
    const int k0   = (lane < 16) ? 0 : 2;
    v2f a2lo = {}, a2hi = {}, a3lo = {}, a3hi = {};
    if (arow < HIDDEN) {
        a2lo.x = w2[arow*8 + k0];     a2lo.y = w2[arow*8 + k0 + 1];
        a2hi.x = w2[arow*8 + k0 + 4]; a2hi.y = w2[arow*8 + k0 + 5];
        a3lo.x = w3[arow*8 + k0];     a3lo.y = w3[arow*8 + k0 + 1];
        a3hi.x = w3[arow*8 + k0 + 4]; a3hi.y = w3[arow*8 + k0 + 5];
    }
    float b2r[8], b3r[8];
    #pragma unroll
    for (int r = 0; r < 8; ++r) { b2r[r] = b2[r]; b3r[r] = b3[r]; }

    // all async tile DMAs must have landed before any wave reads the tile
    asm volatile("s_wait_asynccnt 0" ::: "memory");
    __syncthreads();

    // ---- depthwise conv: wave == channel; lane-per-column, sliding 5-row window ----
    {
        const int c = wave;
        float wk[25];
        #pragma unroll
        for (int k = 0; k < 25; ++k) wk[k] = s_w[c*25 + k];
        const float  b1c = bias1[c];
        const float* P   = s_psi + c * PSI_PLANE;

        #pragma unroll
        for (int s = 0; s < 2; ++s) {
            const int col = s * 32 + lane;          // output column 0..63
            float win[25];                          // win[slot*5+dc], slot = row % 5
            #pragma unroll
            for (int ri = 0; ri < 4; ++ri)
                #pragma unroll
                for (int dc = 0; dc < 5; ++dc)
                    win[ri*5 + dc] = P[ri*LCOLS + col + (COL0 - 2) + dc];
            #pragma unroll
            for (int r = 0; r < TILE_R; ++r) {
                const int snew = (r + 4) % 5;
                #pragma unroll
                for (int dc = 0; dc < 5; ++dc)
                    win[snew*5 + dc] = P[(r + 4)*LCOLS + col + (COL0 - 2) + dc];
                float acc = 0.0f;
                #pragma unroll
                for (int dr = 0; dr < 5; ++dr) {
                    const int sl = (r + dr) % 5;
                    #pragma unroll
                    for (int dc = 0; dc < 5; ++dc)
                        acc = fmaf(wk[dr*5 + dc], win[sl*5 + dc], acc);
                }
                s_z1[c * Z1_PLANE + r * RESOL + col] = leaky(acc + b1c);
            }
        }
    }

    __syncthreads();

    // ---- pointwise 8x8 GEMMs via v_wmma_f32_16x16x4_f32, 16 pixels/group ----
    const int p = lane & 15;
    #pragma unroll
    for (int i = 0; i < 4; ++i) {
        const int g    = wave * 4 + i;     // 32 groups of 16 pixels over the tile
        const int row  = g >> 2;
        const int colb = (g & 3) * 16;
        const int pxb  = row * RESOL + colb;

        // B fragment: lanes 0-15 hold K={0,1}, lanes 16-31 K={2,3} of pixel p
        v2f blo, bhi;
        blo.x = s_z1[(k0    ) * Z1_PLANE + pxb + p];
        blo.y = s_z1[(k0 + 1) * Z1_PLANE + pxb + p];
        bhi.x = s_z1[(k0 + 4) * Z1_PLANE + pxb + p];
        bhi.y = s_z1[(k0 + 5) * Z1_PLANE + pxb + p];

        // layer 2: C pre-loaded with bias (D row r lives in VGPR r, lanes 0-15)
        v8f c2;
        #pragma unroll
        for (int r = 0; r < 8; ++r) c2[r] = b2r[r];
        c2 = __builtin_amdgcn_wmma_f32_16x16x4_f32(false, a2lo, false, blo,
                                                   (short)0, c2, false, false);
        c2 = __builtin_amdgcn_wmma_f32_16x16x4_f32(false, a2hi, false, bhi,
                                                   (short)0, c2, false, false);

        float z2[8];
        #pragma unroll
        for (int r = 0; r < 8; ++r) z2[r] = leaky(c2[r]);

        // repack D (channel-per-VGPR at lanes 0-15) into next B-fragment layout
        float t[8];
        #pragma unroll
        for (int r = 0; r < 8; ++r) t[r] = __shfl(z2[r], p, 32);
        v2f b3lo, b3hi;
        b3lo.x = (lane < 16) ? t[0] : t[2];
        b3lo.y = (lane < 16) ? t[1] : t[3];
        b3hi.x = (lane < 16) ? t[4] : t[6];
        b3hi.y = (lane < 16) ? t[5] : t[7];

        // layer 3: C pre-loaded with bias + psi residual (from LDS tile)
        v8f c3;
        #pragma unroll
        for (int r = 0; r < 8; ++r)
            c3[r] = b3r[r] + s_psi[r * PSI_PLANE + (row + 2) * LCOLS + (colb + p + COL0)];
        c3 = __builtin_amdgcn_wmma_f32_16x16x4_f32(false, a3lo, false, b3lo,
                                                   (short)0, c3, false, false);
        c3 = __builtin_amdgcn_wmma_f32_16x16x4_f32(false, a3hi, false, b3hi,
                                                   (short)0, c3, false, false);

        if (lane < 16) {
            const int grow = r0 + row, gcol = colb + p;
            #pragma unroll
            for (int r = 0; r < 8; ++r)
                out[((b * HIDDEN + r) * RESOL + grow) * RESOL + gcol] = tanhf(c3[r]);
        }
    }
}

extern "C" void kernel_launch(void* const* d_in, const int* in_sizes, int n_in,
                              void* d_out, int out_size, void* d_ws, size_t ws_size,
                              hipStream_t stream) {
    const float* psi     = (const float*)d_in[0];
    const float* filter1 = (const float*)d_in[1];
    const float* bias1   = (const float*)d_in[2];
    const float* w2      = (const float*)d_in[3];
    const float* b2      = (const float*)d_in[4];
    const float* w3      = (const float*)d_in[5];
    const float* b3      = (const float*)d_in[6];
    float* out           = (float*)d_out;

    const int bs = in_sizes[0] / (HIDDEN * RESOL * RESOL);   // 2048
    dim3 grid(RESOL / TILE_R, bs);                           // 8 x 2048 workgroups
    ca_step_kernel<<<grid, 256, 0, stream>>>(psi, filter1, bias1, w2, b2, w3, b3, out);
}